// PatchIoULoss_72980084293836
// MI455X (gfx1250) — compile-verified
//
#include <hip/hip_runtime.h>

// ---------------------------------------------------------------------------
// PatchIoULoss for MI455X (gfx1250, wave32).
// Only the (16,1,64,64) top-left patch of each (16,1,1024,1024) tensor is
// live: 512 KB total traffic -> latency-bound. Strategy:
//   kernel 1: 64 blocks (4 per batch) x 256 threads, one float4 pair per
//             thread (coalesced 256B row segments), LDS funnel to wave 0,
//             final 32-lane sum via V_WMMA_F32_16X16X4_F32 (ones-matrix
//             reduction trick). Partials (P, T, PT) -> d_ws.
//   kernel 2: one wave combines 64x3 partials, does the 16 divisions and
//             the batch sum deterministically, writes scalar d_out.
// ---------------------------------------------------------------------------

typedef float v2f __attribute__((ext_vector_type(2)));
typedef float v8f __attribute__((ext_vector_type(8)));

#define IMG        (1024 * 1024)   // elements per batch image (C=1)
#define ROW_F4     256             // float4s per image row (1024 floats)
#define NBATCH     16
#define SUBS       4               // blocks per batch
#define NBLK       (NBATCH * SUBS) // 64

// Sum v over all 32 lanes of the wave using one f32 WMMA.
// A (16x4) is fed v in both A-VGPRs (every A slot = some lane's v, each lane
// contributes exactly 2 slots); B (4x16) = all ones. Then
//   D[m,n] = sum_k A[m,k]  and  sum_m D[m,n] = 2 * sum_lanes(v).
// D layout (documented): VGPR j holds D[j,n] in lanes 0-15 and D[j+8,n] in
// lanes 16-31, so (sum of 8 D VGPRs) + xor-16 shuffle = 2 * total.
// EXEC must be all ones: call only from a fully-active wave.
__device__ __forceinline__ float wave32_sum_wmma(float v) {
    v2f a = {v, v};
    v2f b = {1.0f, 1.0f};
    v8f c = {};
    v8f d = __builtin_amdgcn_wmma_f32_16x16x4_f32(
        /*neg_a=*/false, a, /*neg_b=*/false, b,
        /*c_mod=*/(short)0, c, /*reuse_a=*/false, /*reuse_b=*/false);
    float s = d[0] + d[1] + d[2] + d[3] + d[4] + d[5] + d[6] + d[7];
    s += __shfl_xor(s, 16, 32);
    return 0.5f * s;
}

__global__ void __launch_bounds__(256)
patch_iou_partial(const float* __restrict__ pred,
                  const float* __restrict__ target,
                  float* __restrict__ ws) {
    const int blk  = blockIdx.x;      // [0,64)
    const int bat  = blk >> 2;        // batch index
    const int sub  = blk & 3;         // 16-row slab within the 64-row patch
    const int tid  = threadIdx.x;     // [0,256)

    // One float4 of pred + one of target per thread, coalesced:
    // tid 0..15 cover one 256-byte row segment.
    const int row_local = tid >> 4;             // [0,16)
    const int c4        = tid & 15;             // [0,16) float4s = 64 cols
    const int row       = sub * 16 + row_local; // [0,64)
    const size_t f4idx  = (size_t)bat * (IMG / 4) + (size_t)row * ROW_F4 + c4;

    const float4 p = reinterpret_cast<const float4*>(pred)[f4idx];
    const float4 t = reinterpret_cast<const float4*>(target)[f4idx];

    float sp  = p.x + p.y + p.z + p.w;
    float st  = t.x + t.y + t.z + t.w;
    float spt = p.x * t.x + p.y * t.y + p.z * t.z + p.w * t.w;

    __shared__ float red[3][256];
    red[0][tid] = sp;
    red[1][tid] = st;
    red[2][tid] = spt;
    __syncthreads();

    if (tid < 32) {  // wave 0, fully active (EXEC all-ones for WMMA)
        float a0 = 0.f, a1 = 0.f, a2 = 0.f;
#pragma unroll
        for (int i = 0; i < 8; ++i) {
            a0 += red[0][tid + 32 * i];
            a1 += red[1][tid + 32 * i];
            a2 += red[2][tid + 32 * i];
        }
        const float P  = wave32_sum_wmma(a0);
        const float T  = wave32_sum_wmma(a1);
        const float PT = wave32_sum_wmma(a2);
        if (tid == 0) {
            ws[blk]            = P;
            ws[NBLK + blk]     = T;
            ws[2 * NBLK + blk] = PT;
        }
    }
}

__global__ void __launch_bounds__(32)
patch_iou_finalize(const float* __restrict__ ws, float* __restrict__ out) {
    const int lane = threadIdx.x;  // one wave
    float loss = 0.0f;
    if (lane < NBATCH) {
        float P = 0.f, T = 0.f, PT = 0.f;
#pragma unroll
        for (int s = 0; s < SUBS; ++s) {
            const int blk = lane * SUBS + s;
            P  += ws[blk];
            T  += ws[NBLK + blk];
            PT += ws[2 * NBLK + blk];
        }
        loss = 1.0f - PT / (P + T - PT);
    }
#pragma unroll
    for (int o = 16; o >= 1; o >>= 1)
        loss += __shfl_xor(loss, o, 32);
    if (lane == 0)
        out[0] = loss;
}

extern "C" void kernel_launch(void* const* d_in, const int* in_sizes, int n_in,
                              void* d_out, int out_size, void* d_ws, size_t ws_size,
                              hipStream_t stream) {
    const float* pred   = (const float*)d_in[0];
    const float* target = (const float*)d_in[1];
    float*       ws     = (float*)d_ws;     // needs 3*64*4 = 768 bytes
    float*       out    = (float*)d_out;

    patch_iou_partial<<<NBLK, 256, 0, stream>>>(pred, target, ws);
    patch_iou_finalize<<<1, 32, 0, stream>>>(ws, out);
}